// PhysicsGuidedEncoder_25967372272024
// MI455X (gfx1250) — compile-verified
//
#include <hip/hip_runtime.h>
#include <math.h>

#define HID 128
#define LDSTRIDE 132   // 128 + 4: bank-conflict-free for 16-row strided A-frag reads

typedef float v2f __attribute__((ext_vector_type(2)));
typedef float v8f __attribute__((ext_vector_type(8)));

// One wave computes a 16x16 f32 tile of  A(16x128, LDS) * B(128x128) using
// V_WMMA_F32_16X16X4_F32 (fp32-exact), K swept 4 at a time.
// B is pre-packed K-pair-interleaved:  Bp[k>>1][2c + (k&1)]  ->  the (k,k+1)
// pair each lane needs is one 8-byte global_load_b64, 128B contiguous per
// half-wave (coalesced, WGP$/L2 resident).
// A-frag layout (ISA 7.12.2): lanes 0-15 -> K = k0,k0+1 ; lanes 16-31 -> K = k0+2,k0+3.
__device__ __forceinline__ v8f wave_gemm_16x128x16(const float* Alds,
                                                   const v2f* __restrict__ Bp,
                                                   int n0, int lane) {
  v8f c = {0.f, 0.f, 0.f, 0.f, 0.f, 0.f, 0.f, 0.f};
  const int row = lane & 15;
  const int kh  = (lane >> 4) << 1;   // 0 or 2
  const int col = lane & 15;
#pragma unroll
  for (int k0 = 0; k0 < HID; k0 += 4) {
    v2f a, b;
    a.x = Alds[row * LDSTRIDE + k0 + kh];
    a.y = Alds[row * LDSTRIDE + k0 + kh + 1];
    b   = Bp[((k0 + kh) >> 1) * HID + n0 + col];   // one b64 load
    c = __builtin_amdgcn_wmma_f32_16x16x4_f32(false, a, false, b, (short)0, c,
                                              false, false);
  }
  return c;
}

// Pack W[k][c] (128x128 row-major) -> P[k>>1][2c+(k&1)]; all 2L matrices.
__global__ void pack_pairs_k(const float* __restrict__ Wnode,
                             const float* __restrict__ Wedge,
                             float* __restrict__ bp) {
  const int m = blockIdx.x;            // [0, 2L): even = lnode[l], odd = ledge[l]
  const int l = m >> 1;
  const float* srcW =
      (m & 1) ? (Wedge + (size_t)l * HID * HID) : (Wnode + (size_t)l * HID * HID);
  float* dstW = bp + (size_t)m * HID * HID;
  for (int e = threadIdx.x; e < HID * HID; e += blockDim.x) {
    int k = e >> 7, c = e & 127;
    dstW[(k >> 1) * 256 + c * 2 + (k & 1)] = srcW[e];
  }
}

// h[n] = x[n] @ node_w + node_b   (K=16 -> plain FMA, memory bound)
__global__ void embed_nodes_k(const float* __restrict__ x,
                              const float* __restrict__ w,
                              const float* __restrict__ b,
                              float* __restrict__ h) {
  const int node = blockIdx.x;
  const int c = threadIdx.x;           // 128 threads
  float acc = b[c];
#pragma unroll
  for (int k = 0; k < 16; ++k) acc = fmaf(x[node * 16 + k], w[k * HID + c], acc);
  h[(size_t)node * HID + c] = acc;
}

// hn = h @ W + bias  via WMMA, 16 nodes per wave, 4 waves per block.
__global__ void node_linear_k(const float* __restrict__ h,
                              const float* __restrict__ Wp,   // pair-packed
                              const float* __restrict__ bias,
                              float* __restrict__ hn, int nNodes) {
  __shared__ float hl[4][16 * LDSTRIDE];
  const int w    = threadIdx.x >> 5;
  const int lane = threadIdx.x & 31;
  const int tile = blockIdx.x * 4 + w;
  if (tile * 16 >= nNodes) return;     // wave-uniform; EXEC stays all-ones for WMMA
  const int n0base = tile * 16;

  // stage 16x128 tile into LDS (scalar stores keep the 132-stride alignment simple)
#pragma unroll
  for (int i = 0; i < 16; ++i) {
    int flat = i * 32 + lane;          // 512 float4s
    int row = flat >> 5, q = flat & 31;
    float4 v = *(const float4*)(h + (size_t)(n0base + row) * HID + q * 4);
    float* d = &hl[w][row * LDSTRIDE + q * 4];
    d[0] = v.x; d[1] = v.y; d[2] = v.z; d[3] = v.w;
  }

  const int half = lane >> 4;
  for (int n0 = 0; n0 < HID; n0 += 16) {
    v8f c = wave_gemm_16x128x16(&hl[w][0], (const v2f*)Wp, n0, lane);
    const int col = n0 + (lane & 15);
    const float bb = bias[col];
#pragma unroll
    for (int j = 0; j < 8; ++j) {
      int row = j + (half << 3);
      hn[(size_t)(n0base + row) * HID + col] = c[j] + bb;
    }
  }
}

// Fused per-edge kernel: recompute ea = edge_attr@edge_w+edge_b (L2-hot, 8-wide K),
// gate y = sigmoid(ea . adm_w + adm_b), e = ea @ ledge_w + ledge_b via WMMA,
// msg = y*(hn[src]+e), atomic scatter-add into agg[dst].
__global__ void edge_msg_k(const float* __restrict__ edge_attr,
                           const int* __restrict__ src,
                           const int* __restrict__ dst,
                           const float* __restrict__ edge_w,
                           const float* __restrict__ edge_b,
                           const float* __restrict__ ledge_wp,  // pair-packed
                           const float* __restrict__ ledge_b,
                           const float* __restrict__ adm_w,
                           const float* __restrict__ adm_b,
                           const float* __restrict__ hn,
                           float* __restrict__ agg, int nE) {
  __shared__ float ealds[4][16 * LDSTRIDE];
  __shared__ float attl[4][16 * 8];
  __shared__ float yl[4][16];
  __shared__ int   sl[4][16];
  __shared__ int   dl[4][16];

  const int w    = threadIdx.x >> 5;
  const int lane = threadIdx.x & 31;
  const int tile = blockIdx.x * 4 + w;
  const int e0   = tile * 16;
  if (e0 >= nE) return;                // wave-uniform

  // --- stage edge_attr 16x8 (one float4 per lane) + indices ---
  {
    float4 v = *(const float4*)(edge_attr + (size_t)e0 * 8 + lane * 4);
    *(float4*)(&attl[w][lane * 4]) = v;
  }
  if (lane < 16) {
    sl[w][lane] = src[e0 + lane];
    dl[w][lane] = dst[e0 + lane];
  }

  // --- recompute ea tile (16 x 128) into LDS: each lane owns 4 columns ---
  float wreg[8][4], breg[4];
#pragma unroll
  for (int cc = 0; cc < 4; ++cc) {
    int c = lane + cc * 32;
    breg[cc] = edge_b[c];
#pragma unroll
    for (int k = 0; k < 8; ++k) wreg[k][cc] = edge_w[k * HID + c];
  }
#pragma unroll
  for (int r = 0; r < 16; ++r) {
    float acc[4] = {breg[0], breg[1], breg[2], breg[3]};
#pragma unroll
    for (int k = 0; k < 8; ++k) {
      float a = attl[w][r * 8 + k];    // LDS broadcast (same wave: DS in-order)
#pragma unroll
      for (int cc = 0; cc < 4; ++cc) acc[cc] = fmaf(a, wreg[k][cc], acc[cc]);
    }
#pragma unroll
    for (int cc = 0; cc < 4; ++cc)
      ealds[w][r * LDSTRIDE + lane + cc * 32] = acc[cc];
  }

  // --- admittance gates: y[r] = sigmoid(ea[r] . adm_w + adm_b) ---
  float admreg[4];
#pragma unroll
  for (int cc = 0; cc < 4; ++cc) admreg[cc] = adm_w[lane + cc * 32];
  const float ab = adm_b[0];
#pragma unroll
  for (int r = 0; r < 16; ++r) {
    float p = 0.f;
#pragma unroll
    for (int cc = 0; cc < 4; ++cc)
      p = fmaf(ealds[w][r * LDSTRIDE + lane + cc * 32], admreg[cc], p);
#pragma unroll
    for (int m = 16; m >= 1; m >>= 1) p += __shfl_xor(p, m, 32);  // wave32 reduce
    if (lane == 0) yl[w][r] = 1.f / (1.f + __expf(-(p + ab)));
  }

  // --- WMMA GEMM over 8 column tiles; fuse gate + gather + atomic scatter ---
  const int half = lane >> 4;
  for (int n0 = 0; n0 < HID; n0 += 16) {
    v8f c = wave_gemm_16x128x16(&ealds[w][0], (const v2f*)ledge_wp, n0, lane);
    const int col = n0 + (lane & 15);
    const float lb = ledge_b[col];
#pragma unroll
    for (int j = 0; j < 8; ++j) {
      int r = j + (half << 3);
      float ev  = c[j] + lb;
      float msg = yl[w][r] * (hn[(size_t)sl[w][r] * HID + col] + ev);
      atomicAdd(agg + (size_t)dl[w][r] * HID + col, msg);
    }
  }
}

// h += relu(layernorm(agg) * gamma + beta); one wave per node (4 vals/lane).
__global__ void lnorm_relu_add_k(const float* __restrict__ agg,
                                 const float* __restrict__ gamma,
                                 const float* __restrict__ beta,
                                 float* __restrict__ h, int nNodes) {
  const int node = blockIdx.x * 4 + (threadIdx.x >> 5);
  if (node >= nNodes) return;
  const int lane = threadIdx.x & 31;
  float v[4];
  float s = 0.f;
#pragma unroll
  for (int cc = 0; cc < 4; ++cc) {
    v[cc] = agg[(size_t)node * HID + lane + cc * 32];
    s += v[cc];
  }
#pragma unroll
  for (int m = 16; m >= 1; m >>= 1) s += __shfl_xor(s, m, 32);
  const float mu = s * (1.f / 128.f);
  float var = 0.f;
#pragma unroll
  for (int cc = 0; cc < 4; ++cc) {
    float d = v[cc] - mu;
    var = fmaf(d, d, var);
  }
#pragma unroll
  for (int m = 16; m >= 1; m >>= 1) var += __shfl_xor(var, m, 32);
  const float rinv = rsqrtf(var * (1.f / 128.f) + 1e-5f);
#pragma unroll
  for (int cc = 0; cc < 4; ++cc) {
    int c = lane + cc * 32;
    float xn = (v[cc] - mu) * rinv * gamma[c] + beta[c];
    h[(size_t)node * HID + c] += fmaxf(xn, 0.f);
  }
}

extern "C" void kernel_launch(void* const* d_in, const int* in_sizes, int n_in,
                              void* d_out, int out_size, void* d_ws, size_t ws_size,
                              hipStream_t stream) {
  const float* x       = (const float*)d_in[0];
  const int*   eidx    = (const int*)d_in[1];     // int32 (JAX x64 off)
  const float* eattr   = (const float*)d_in[2];
  const float* node_w  = (const float*)d_in[3];
  const float* node_b  = (const float*)d_in[4];
  const float* edge_w  = (const float*)d_in[5];
  const float* edge_b  = (const float*)d_in[6];
  const float* lnode_w = (const float*)d_in[7];
  const float* lnode_b = (const float*)d_in[8];
  const float* ledge_w = (const float*)d_in[9];
  const float* ledge_b = (const float*)d_in[10];
  const float* adm_w   = (const float*)d_in[11];
  const float* adm_b   = (const float*)d_in[12];
  const float* gamma   = (const float*)d_in[13];
  const float* beta    = (const float*)d_in[14];

  const int N = in_sizes[0] / 16;
  const int E = in_sizes[2] / 8;
  const int L = in_sizes[7] / (HID * HID);

  float* h   = (float*)d_out;                 // h lives in d_out (fully rewritten)
  float* hn  = (float*)d_ws;                  // N*HID
  float* agg = hn + (size_t)N * HID;          // N*HID
  float* bp  = agg + (size_t)N * HID;         // 2L * HID*HID pair-packed weights
  const int* src = eidx;
  const int* dst = eidx + E;

  // Pack all layer weights into K-pair-interleaved layout (once per call).
  pack_pairs_k<<<2 * L, 256, 0, stream>>>(lnode_w, ledge_w, bp);

  embed_nodes_k<<<N, 128, 0, stream>>>(x, node_w, node_b, h);

  const int nodeTiles = (N + 15) / 16;
  const int edgeTiles = (E + 15) / 16;
  const int nlBlocks  = (nodeTiles + 3) / 4;
  const int eBlocks   = (edgeTiles + 3) / 4;
  const int lnBlocks  = (N + 3) / 4;

  for (int l = 0; l < L; ++l) {
    node_linear_k<<<nlBlocks, 128, 0, stream>>>(
        h, bp + (size_t)(2 * l) * HID * HID, lnode_b + (size_t)l * HID, hn, N);
    hipMemsetAsync(agg, 0, (size_t)N * HID * sizeof(float), stream);
    edge_msg_k<<<eBlocks, 128, 0, stream>>>(
        eattr, src, dst, edge_w, edge_b,
        bp + (size_t)(2 * l + 1) * HID * HID, ledge_b + (size_t)l * HID,
        adm_w + (size_t)l * HID, adm_b + l, hn, agg, E);
    lnorm_relu_add_k<<<lnBlocks, 128, 0, stream>>>(
        agg, gamma + (size_t)l * HID, beta + (size_t)l * HID, h, N);
  }
}